// gin_tuple3_Net_67508295958861
// MI455X (gfx1250) — compile-verified
//
#include <hip/hip_runtime.h>

#define N_NODES 50000
#define N_EDGES 600000
#define N_GRAPHS 256
#define D_H 128

typedef float v2f __attribute__((ext_vector_type(2)));
typedef float v8f __attribute__((ext_vector_type(8)));

// ---------------------------------------------------------------------------
// Zero fill (float4-vectorized)
// ---------------------------------------------------------------------------
__global__ __launch_bounds__(256) void zero_f4(float4* __restrict__ dst, int n4) {
    int i = blockIdx.x * 256 + threadIdx.x;
    if (i < n4) dst[i] = make_float4(0.f, 0.f, 0.f, 0.f);
}

// ---------------------------------------------------------------------------
// Scatter-add: S[dst] += X[src] per edge.  One thread = (edge, 4-float chunk).
// X rows (25.6 MB) are L2-resident on MI455X (192 MB L2), so the random
// gather and the atomics both stay on-chip.
// ---------------------------------------------------------------------------
__global__ __launch_bounds__(256) void scatter_add(const float* __restrict__ X,
                                                   const int* __restrict__ ei,
                                                   float* __restrict__ S) {
    int t = blockIdx.x * 256 + threadIdx.x;
    int e = t >> 5;          // edge index
    int c = t & 31;          // 4-float chunk within the 128-wide row
    if (e >= N_EDGES) return;
    int src = ei[e];
    int dst = ei[N_EDGES + e];
    const float4 v = *(const float4*)(X + (size_t)src * D_H + c * 4);
    float* p = S + (size_t)dst * D_H + c * 4;
    atomicAdd(p + 0, v.x);
    atomicAdd(p + 1, v.y);
    atomicAdd(p + 2, v.z);
    atomicAdd(p + 3, v.w);
}

// ---------------------------------------------------------------------------
// Fused GEMM: Y[:, col0:col0+128] = act(X @ W + bias)
//   X: n_rows x K (row stride ldx), W: K x 128, Y row stride ldy.
// 8 waves/block, each wave computes a 16x128 strip with 8 v8f accumulators
// using V_WMMA_F32_16X16X4_F32 (fp32 in, fp32 acc).  W is staged through LDS
// in 128-K chunks, stored K-PAIR INTERLEAVED:
//     sW2[p][n] = { W[kc+2p][n], W[kc+2p+1][n] }   (float2, 64 KB total)
// so the B fragment for (k, half, tile j) is ONE contiguous ds_load_b64 at
// (k/2 + half)*128 + j*16 + lane&15 — it lands directly in the even-aligned
// VGPR pair WMMA wants (no v_mov shuffles, no 2addr loads).
//
// Row indices are clamped (never predicated) so EXEC is all-1s around every
// WMMA per the CDNA5 restriction; only epilogue stores are per-lane guarded.
//
// Operand layouts (ISA 7.12.2, wave32):
//   A 16x4 f32 : lane m in [0,16) holds row m; VGPR v, half h -> K = v + 2h
//   B 4x16 f32 : VGPR v, half h -> K = v + 2h, N = lane&15
//   C/D 16x16  : VGPR r, half h -> M = r + 8h, N = lane&15
// ---------------------------------------------------------------------------
__global__ __launch_bounds__(256) void gemm_bias(const float* __restrict__ X, int ldx, int K,
                                                 const float* __restrict__ W,
                                                 const float* __restrict__ bias,
                                                 float* __restrict__ Y, int ldy, int col0,
                                                 int relu, int n_rows) {
    __shared__ float2 sW2[64 * 128];           // 64 KB of the 320 KB WGP LDS

    const int wave = threadIdx.x >> 5;
    const int lane = threadIdx.x & 31;
    const int half = lane >> 4;
    const int lm   = lane & 15;
    const int row0 = (blockIdx.x * 8 + wave) * 16;

    int rowA = row0 + lm;
    if (rowA > n_rows - 1) rowA = n_rows - 1;  // clamp: keep EXEC all-1s
    const float* xrow = X + (size_t)rowA * ldx + 2 * half;

    v8f acc[8] = {};

    for (int kc = 0; kc < K; kc += 128) {
        __syncthreads();                       // protect previous LDS use
        // Stage 128x128 W chunk into LDS, K-pair interleaved.
        // Task t = (pair p, 4-column group): 2 global b128 loads -> 2 ds b128.
        #pragma unroll 2
        for (int t = threadIdx.x; t < 64 * 32; t += 256) {
            const int pp = t >> 5;
            const int n  = (t & 31) * 4;
            const float4 r0 = *(const float4*)(W + (size_t)(kc + 2 * pp) * 128 + n);
            const float4 r1 = *(const float4*)(W + (size_t)(kc + 2 * pp + 1) * 128 + n);
            float4* dst = (float4*)(sW2 + pp * 128 + n);
            dst[0] = make_float4(r0.x, r1.x, r0.y, r1.y);
            dst[1] = make_float4(r0.z, r1.z, r0.w, r1.w);
        }
        __syncthreads();

        #pragma unroll 4
        for (int k = 0; k < 128; k += 4) {
            // A fragment: this lane's row, K = k + 2h .. k + 2h + 1 (b64 load)
            const v2f a = *(const v2f*)(xrow + kc + k);
            // B fragments: pair index p = k/2 + half, column j*16 + lm
            const float2* bbase = sW2 + (k / 2 + half) * 128 + lm;
            #pragma unroll
            for (int j = 0; j < 8; ++j) {      // 8 column tiles of 16
                const v2f b = *(const v2f*)(bbase + j * 16);
                acc[j] = __builtin_amdgcn_wmma_f32_16x16x4_f32(
                    false, a, false, b, (short)0, acc[j], false, false);
            }
        }
    }

    // Epilogue: bias + optional ReLU, per-lane guarded stores.
    #pragma unroll
    for (int j = 0; j < 8; ++j) {
        const float bv  = bias[j * 16 + lm];
        const int   col = col0 + j * 16 + lm;
        #pragma unroll
        for (int r = 0; r < 8; ++r) {
            const int row = row0 + r + 8 * half;
            float v = acc[j][r] + bv;
            if (relu) v = fmaxf(v, 0.f);
            if (row < n_rows) Y[(size_t)row * ldy + col] = v;
        }
    }
}

// ---------------------------------------------------------------------------
// Global add-pool: G[batch[n]] += H[n].  One thread = (node, 4-float chunk).
// ---------------------------------------------------------------------------
__global__ __launch_bounds__(256) void pool_add(const float* __restrict__ H,
                                                const int* __restrict__ batch,
                                                float* __restrict__ G) {
    int t = blockIdx.x * 256 + threadIdx.x;
    int n = t >> 5;
    int c = t & 31;
    if (n >= N_NODES) return;
    int g = batch[n];
    const float4 v = *(const float4*)(H + (size_t)n * D_H + c * 4);
    float* p = G + (size_t)g * D_H + c * 4;
    atomicAdd(p + 0, v.x);
    atomicAdd(p + 1, v.y);
    atomicAdd(p + 2, v.z);
    atomicAdd(p + 3, v.w);
}

// ---------------------------------------------------------------------------
// Head: out[g] = G[g] . Wl + bl   (256 graphs, one thread each)
// ---------------------------------------------------------------------------
__global__ __launch_bounds__(256) void head(const float* __restrict__ G,
                                            const float* __restrict__ Wl,
                                            const float* __restrict__ bl,
                                            float* __restrict__ out) {
    int i = threadIdx.x;
    float s = bl[0];
    const float* g = G + (size_t)i * D_H;
    #pragma unroll 8
    for (int k = 0; k < D_H; ++k) s += g[k] * Wl[k];
    out[i] = s;
}

// ---------------------------------------------------------------------------
// Orchestration
// ---------------------------------------------------------------------------
extern "C" void kernel_launch(void* const* d_in, const int* in_sizes, int n_in,
                              void* d_out, int out_size, void* d_ws, size_t ws_size,
                              hipStream_t stream) {
    (void)in_sizes; (void)n_in; (void)out_size; (void)ws_size;

    const float* x     = (const float*)d_in[0];
    const int*   ei[3] = {(const int*)d_in[1], (const int*)d_in[2], (const int*)d_in[3]};
    const int*   batch = (const int*)d_in[4];
    const float* p[34];
    for (int i = 0; i < 34; ++i) p[i] = (const float*)d_in[5 + i];
    // p layout: conv_1_b -> p[4b..4b+3] (W1,b1,W2,b2); mlp_1 -> p[12..15];
    //           conv_2_b -> p[16+4b..]; mlp_2 -> p[28..31]; lin -> p[32],p[33]

    const size_t NF = (size_t)N_NODES * D_H;       // 6.4M floats
    float* S    = (float*)d_ws;                    // x + agg scratch
    float* T    = S + NF;                          // hidden scratch
    float* Hcat = T + NF;                          // 50000 x 384 concat
    float* h    = Hcat + (size_t)N_NODES * 384;    // layer output
    float* G    = h + NF;                          // 256 x 128 pooled

    const int edge_blocks = (N_EDGES * 32 + 255) / 256;
    const int gemm_blocks = (N_NODES + 127) / 128;
    const int pool_blocks = (N_NODES * 32 + 255) / 256;

    // ---- layer 1: three GIN branches -> Hcat ----
    for (int b = 0; b < 3; ++b) {
        hipMemcpyAsync(S, x, NF * sizeof(float), hipMemcpyDeviceToDevice, stream);
        scatter_add<<<edge_blocks, 256, 0, stream>>>(x, ei[b], S);
        gemm_bias<<<gemm_blocks, 256, 0, stream>>>(S, 128, 128, p[4*b+0], p[4*b+1],
                                                   T, 128, 0, /*relu=*/1, N_NODES);
        gemm_bias<<<gemm_blocks, 256, 0, stream>>>(T, 128, 128, p[4*b+2], p[4*b+3],
                                                   Hcat, 384, b * 128, /*relu=*/0, N_NODES);
    }
    // mlp_1 over the 384-wide concat
    gemm_bias<<<gemm_blocks, 256, 0, stream>>>(Hcat, 384, 384, p[12], p[13],
                                               T, 128, 0, 1, N_NODES);
    gemm_bias<<<gemm_blocks, 256, 0, stream>>>(T, 128, 128, p[14], p[15],
                                               h, 128, 0, 0, N_NODES);

    // ---- layer 2: three GIN branches (ReLU on branch output) -> Hcat ----
    for (int b = 0; b < 3; ++b) {
        hipMemcpyAsync(S, h, NF * sizeof(float), hipMemcpyDeviceToDevice, stream);
        scatter_add<<<edge_blocks, 256, 0, stream>>>(h, ei[b], S);
        gemm_bias<<<gemm_blocks, 256, 0, stream>>>(S, 128, 128, p[16+4*b+0], p[16+4*b+1],
                                                   T, 128, 0, 1, N_NODES);
        gemm_bias<<<gemm_blocks, 256, 0, stream>>>(T, 128, 128, p[16+4*b+2], p[16+4*b+3],
                                                   Hcat, 384, b * 128, /*relu=*/1, N_NODES);
    }
    // mlp_2; final node features written into S (free by now)
    gemm_bias<<<gemm_blocks, 256, 0, stream>>>(Hcat, 384, 384, p[28], p[29],
                                               T, 128, 0, 1, N_NODES);
    gemm_bias<<<gemm_blocks, 256, 0, stream>>>(T, 128, 128, p[30], p[31],
                                               S, 128, 0, 0, N_NODES);

    // ---- global add pool + linear head ----
    zero_f4<<<(N_GRAPHS * D_H / 4 + 255) / 256, 256, 0, stream>>>((float4*)G,
                                                                  N_GRAPHS * D_H / 4);
    pool_add<<<pool_blocks, 256, 0, stream>>>(S, batch, G);
    head<<<1, 256, 0, stream>>>(G, p[32], p[33], (float*)d_out);
}